// FiniteScalarQuantizer_24635932410453
// MI455X (gfx1250) — compile-verified
//
#include <hip/hip_runtime.h>

#ifndef __has_builtin
#define __has_builtin(x) 0
#endif

#define DDIM   64
#define NBINS  256
#define NELEM  4194304          // 65536*64
#define BLOCK  256
#define GRID   1024
#define ELEMS_PER_BLOCK (NELEM / GRID)               // 4096
#define VEC4_PER_THREAD (ELEMS_PER_BLOCK / 4 / BLOCK) // 4

typedef __attribute__((ext_vector_type(4))) int v4i;
typedef __attribute__((address_space(1))) v4i* gv4i_ptr; // global int4*
typedef __attribute__((address_space(3))) v4i* lv4i_ptr; // LDS int4*

__global__ __launch_bounds__(BLOCK) void
fsq_quantize_kernel(const float* __restrict__ z,
                    const float* __restrict__ bins,
                    float* __restrict__ out_zq,
                    float* __restrict__ out_idx,
                    float* __restrict__ partials)
{
    __shared__ float s_bins[DDIM * NBINS];   // 64 KB codebook in LDS

    const int tid = threadIdx.x;

    // ---- Stage 64x256 f32 codebook into LDS (CDNA5 async-to-LDS path) ----
#if __has_builtin(__builtin_amdgcn_global_load_async_to_lds_b128)
    {
        gv4i_ptr gsrc = (gv4i_ptr)bins;      // cast away const; int4 units
        lv4i_ptr ldst = (lv4i_ptr)s_bins;
        #pragma unroll
        for (int i = 0; i < (DDIM * NBINS) / (BLOCK * 4); ++i) { // 16 iters
            const int k = tid + i * BLOCK;   // int4 index (16B granules)
            __builtin_amdgcn_global_load_async_to_lds_b128(gsrc + k, ldst + k, 0, 0);
        }
    }
  #if __has_builtin(__builtin_amdgcn_s_wait_asynccnt)
    __builtin_amdgcn_s_wait_asynccnt(0);
  #else
    asm volatile("s_wait_asynccnt 0" ::: "memory");
  #endif
#else
    #pragma unroll
    for (int i = 0; i < (DDIM * NBINS) / (BLOCK * 4); ++i) {
        const int q = tid + i * BLOCK;
        ((float4*)s_bins)[q] = ((const float4*)bins)[q];
    }
#endif
    __syncthreads();

    // Each thread always touches the same 4 dims: dbase = (4*tid) & 63,
    // because the per-iteration stride (1024 elems) is a multiple of DDIM.
    const int dbase = (tid * 4) & (DDIM - 1);
    float b0[4], inv[4];
    #pragma unroll
    for (int j = 0; j < 4; ++j) {
        const int d   = dbase + j;
        const float lo  = s_bins[d * NBINS];
        const float hi  = s_bins[d * NBINS + NBINS - 1];
        const float stp = (hi - lo) * (1.0f / (NBINS - 1));
        b0[j]  = lo;
        inv[j] = (stp != 0.0f) ? (1.0f / stp) : 0.0f;
    }

    float acc = 0.0f;
    const int block_base = blockIdx.x * ELEMS_PER_BLOCK;   // element index
    #pragma unroll
    for (int it = 0; it < VEC4_PER_THREAD; ++it) {
        const int q = tid + it * BLOCK;                    // float4 slot
        const int e = block_base + q * 4;                  // element index
        const float4 v = reinterpret_cast<const float4*>(z)[(block_base >> 2) + q];
        const float zv[4] = {v.x, v.y, v.z, v.w};
        #pragma unroll
        for (int j = 0; j < 4; ++j) {
            const int d  = dbase + j;
            const float zz = zv[j];
            // closed-form nearest bin on a uniform grid
            int i0 = (int)rintf((zz - b0[j]) * inv[j]);
            i0 = i0 < 0 ? 0 : (i0 > NBINS - 1 ? NBINS - 1 : i0);
            const int im = (i0 > 0)         ? i0 - 1 : i0;
            const int ip = (i0 < NBINS - 1) ? i0 + 1 : i0;
            // bit-exact codebook values from LDS, +/-1 refinement
            const float c  = s_bins[d * NBINS + i0];
            const float cm = s_bins[d * NBINS + im];
            const float cp = s_bins[d * NBINS + ip];
            float bd = fabsf(zz - c);
            int   bi = i0;
            float bv = c;
            const float dm = fabsf(zz - cm);
            if (dm <= bd) { bd = dm; bi = im; bv = cm; }  // tie -> lower index
            const float dp = fabsf(zz - cp);
            if (dp <  bd) { bd = dp; bi = ip; bv = cp; }
            const float diff = zz - bv;
            acc = fmaf(diff, diff, acc);
            out_zq[e + j]  = bv;          // z_q == zq numerically (STE)
            out_idx[e + j] = (float)bi;   // indices, flat in output dtype
        }
    }

    // ---- deterministic per-block partial sum (wave32 shuffle + LDS) ------
    #pragma unroll
    for (int off = 16; off > 0; off >>= 1)
        acc += __shfl_down(acc, off, 32);
    __syncthreads();                       // bins reads done; reuse LDS
    if ((tid & 31) == 0) s_bins[tid >> 5] = acc;
    __syncthreads();
    if (tid == 0) {
        float t = 0.0f;
        #pragma unroll
        for (int w = 0; w < BLOCK / 32; ++w) t += s_bins[w];
        partials[blockIdx.x] = t;
    }
}

__global__ __launch_bounds__(BLOCK) void
fsq_reduce_kernel(const float* __restrict__ partials, float* __restrict__ out_loss)
{
    __shared__ float s[BLOCK / 32];
    float a = 0.0f;
    for (int i = threadIdx.x; i < GRID; i += BLOCK) a += partials[i];
    #pragma unroll
    for (int off = 16; off > 0; off >>= 1)
        a += __shfl_down(a, off, 32);
    if ((threadIdx.x & 31) == 0) s[threadIdx.x >> 5] = a;
    __syncthreads();
    if (threadIdx.x == 0) {
        float t = 0.0f;
        #pragma unroll
        for (int w = 0; w < BLOCK / 32; ++w) t += s[w];
        // fsq_loss = commitment + BETA*codebook = 2 * mean(diff^2)
        out_loss[0] = 2.0f * t * (1.0f / (float)NELEM);
    }
}

extern "C" void kernel_launch(void* const* d_in, const int* in_sizes, int n_in,
                              void* d_out, int out_size, void* d_ws, size_t ws_size,
                              hipStream_t stream)
{
    const float* z    = (const float*)d_in[0];   // 4M f32, last dim 64
    const float* bins = (const float*)d_in[1];   // (64,256) f32
    float* out      = (float*)d_out;             // [loss][z_q 4M][idx 4M]
    float* out_loss = out;
    float* out_zq   = out + 1;
    float* out_idx  = out + 1 + NELEM;
    float* partials = (float*)d_ws;              // GRID floats of scratch

    fsq_quantize_kernel<<<GRID, BLOCK, 0, stream>>>(z, bins, out_zq, out_idx, partials);
    fsq_reduce_kernel<<<1, BLOCK, 0, stream>>>(partials, out_loss);
}